// ChebyshevGCNN_41996190220765
// MI455X (gfx1250) — compile-verified
//
#include <hip/hip_runtime.h>

// Problem constants (from reference)
#define BB   16
#define NN   4096
#define EE   65536
#define CIN  128
#define COUT 128
// MAX_DEGREE = 3 -> 4 Chebyshev terms

typedef __attribute__((ext_vector_type(2))) float v2f;
typedef __attribute__((ext_vector_type(8))) float v8f;

// ---------------------------------------------------------------------------
// dst = scale * src   (src == nullptr -> dst = 0).  float4 vectorized.
// ---------------------------------------------------------------------------
__global__ void scale_copy_kernel(float* __restrict__ dst,
                                  const float* __restrict__ src,
                                  float scale, int n4) {
    int i = blockIdx.x * blockDim.x + threadIdx.x;
    if (i >= n4) return;
    float4 v;
    if (src) {
        float4 s = ((const float4*)src)[i];
        v = make_float4(scale * s.x, scale * s.y, scale * s.z, scale * s.w);
    } else {
        v = make_float4(0.f, 0.f, 0.f, 0.f);
    }
    ((float4*)dst)[i] = v;
}

// ---------------------------------------------------------------------------
// SpMM scatter:  y[b, rows[e], :] += scale * vals[e] * x[b, cols[e], :]
// One wave (32 lanes) per edge; each lane owns 4 channels (float4).
// Edge metadata addresses are wave-uniform -> scalar loads.
// ---------------------------------------------------------------------------
__global__ void spmm_scatter_kernel(float* __restrict__ y,
                                    const float* __restrict__ x,
                                    const float* __restrict__ vals,
                                    const int* __restrict__ rows,
                                    const int* __restrict__ cols,
                                    float scale) {
    const int gwave = (blockIdx.x * blockDim.x + threadIdx.x) >> 5; // edge idx over B*E
    if (gwave >= BB * EE) return;
    const int b = gwave >> 16;        // E == 65536
    const int r = rows[gwave];
    const int c = cols[gwave];
    const float v = vals[gwave] * scale;
    const int lane = threadIdx.x & 31;

    const float4 xv = ((const float4*)(x + ((size_t)b * NN + c) * CIN))[lane];
    float* yp = y + ((size_t)b * NN + r) * CIN + lane * 4;
    atomicAdd(yp + 0, v * xv.x);
    atomicAdd(yp + 1, v * xv.y);
    atomicAdd(yp + 2, v * xv.z);
    atomicAdd(yp + 3, v * xv.w);
}

// ---------------------------------------------------------------------------
// Fused Chebyshev GEMM:
//   out = relu(bias + T0@W0 + T1@W1 + T2@W2 + T3@W3)
// T* are (B*N, 128) row-major fp32, W is (4,128,128) row-major fp32.
// Block: 512 threads = 16 waves = 2 (M) x 8 (N) tiles of 16x16.
// Block output tile: 32 rows x 128 cols.  A staged in LDS (stride 132 pad),
// W fragments read from global (hot in L2: 256 KB total).
// WMMA: v_wmma_f32_16x16x4_f32, 32 K-steps per term, exact fp32.
// ---------------------------------------------------------------------------
__global__ __launch_bounds__(512)
void cheb_gemm_kernel(const float* __restrict__ t0,
                      const float* __restrict__ t1,
                      const float* __restrict__ t2,
                      const float* __restrict__ t3,
                      const float* __restrict__ weights,
                      const float* __restrict__ bias,
                      float* __restrict__ out) {
    __shared__ float As[32 * 132];   // 32 rows x 128 K, padded stride 132 (bank-conflict free)

    const float* terms[4] = {t0, t1, t2, t3};

    const int tid  = threadIdx.x;
    const int wave = tid >> 5;
    const int lane = tid & 31;
    const int wm   = wave >> 3;      // 0..1   M tile within block
    const int wn   = wave & 7;       // 0..7   N tile within block
    const int l    = lane & 15;
    const int half = lane >> 4;

    const size_t row0 = (size_t)blockIdx.x * 32;

    const int arow = wm * 16 + l;            // A row within block tile
    const int bn   = wn * 16 + l;            // B column (output channel)

    v8f acc = {};

#pragma unroll
    for (int t = 0; t < 4; ++t) {
        __syncthreads();
        // Stage 32x128 fp32 tile of term t into LDS: 1024 float4, 512 threads.
        const float* src = terms[t] + row0 * CIN;
#pragma unroll
        for (int i = tid; i < 1024; i += 512) {
            const int r  = i >> 5;           // 0..31 row
            const int cv = i & 31;           // 0..31 float4 within row
            const float4 v = ((const float4*)(src + r * CIN))[cv];
            float* d = &As[r * 132 + cv * 4];
            d[0] = v.x; d[1] = v.y; d[2] = v.z; d[3] = v.w;
        }
        __syncthreads();

        const float* W = weights + t * CIN * COUT;
#pragma unroll 4
        for (int k = 0; k < 128; k += 4) {
            const int kk = k + 2 * half;
            // A fragment (16x4 f32): VGPR0 = K=kk, VGPR1 = K=kk+1 (per half-wave)
            v2f a;
            a.x = As[arow * 132 + kk];
            a.y = As[arow * 132 + kk + 1];
            // B fragment (4x16 f32): VGPR0 = K-row kk, VGPR1 = K-row kk+1
            v2f bf;
            bf.x = W[kk * COUT + bn];
            bf.y = W[(kk + 1) * COUT + bn];
            acc = __builtin_amdgcn_wmma_f32_16x16x4_f32(
                /*neg_a=*/false, a, /*neg_b=*/false, bf,
                /*c_mod=*/(short)0, acc, /*reuse_a=*/false, /*reuse_b=*/false);
        }
    }

    // Epilogue: bias + ReLU.  C/D layout: VGPR r -> M = r + 8*half, N = l.
    const float bv = bias[bn];
    float* op = out + (row0 + (size_t)wm * 16 + 8 * half) * COUT + bn;
#pragma unroll
    for (int r = 0; r < 8; ++r) {
        const float v = acc[r] + bv;
        op[(size_t)r * COUT] = fmaxf(v, 0.f);
    }
}

// ---------------------------------------------------------------------------
// Launch sequence (all on `stream`, graph-capture safe):
//   T1 = 0;              T1 += vals * x[cols]          (cheb1 = L x)
//   T2 = -x;             T2 += 2*vals*T1[cols]         (cheb2 = 2 L T1 - T0)
//   T3 = -T1;            T3 += 2*vals*T2[cols]         (cheb3 = 2 L T2 - T1)
//   out = relu(bias + x@W0 + T1@W1 + T2@W2 + T3@W3)    (WMMA fused GEMM)
// Workspace: 3 * B*N*CIN fp32 = 96 MB.
// ---------------------------------------------------------------------------
extern "C" void kernel_launch(void* const* d_in, const int* in_sizes, int n_in,
                              void* d_out, int out_size, void* d_ws, size_t ws_size,
                              hipStream_t stream) {
    const float* x       = (const float*)d_in[0];
    const float* vals    = (const float*)d_in[1];
    const float* weights = (const float*)d_in[2];
    const float* bias    = (const float*)d_in[3];
    const int*   rows    = (const int*)d_in[4];
    const int*   cols    = (const int*)d_in[5];
    float* out = (float*)d_out;

    const size_t feat = (size_t)BB * NN * CIN;   // 8,388,608 floats (32 MB)
    float* T1 = (float*)d_ws;
    float* T2 = T1 + feat;
    float* T3 = T2 + feat;

    const int n4       = (int)(feat / 4);
    const int initGrid = (n4 + 255) / 256;
    const int spmmGrid = (BB * EE * 32) / 256;   // one wave per edge

    // T1 = spmm(x)
    scale_copy_kernel<<<initGrid, 256, 0, stream>>>(T1, nullptr, 0.f, n4);
    spmm_scatter_kernel<<<spmmGrid, 256, 0, stream>>>(T1, x, vals, rows, cols, 1.0f);
    // T2 = 2*spmm(T1) - x
    scale_copy_kernel<<<initGrid, 256, 0, stream>>>(T2, x, -1.f, n4);
    spmm_scatter_kernel<<<spmmGrid, 256, 0, stream>>>(T2, T1, vals, rows, cols, 2.0f);
    // T3 = 2*spmm(T2) - T1
    scale_copy_kernel<<<initGrid, 256, 0, stream>>>(T3, T1, -1.f, n4);
    spmm_scatter_kernel<<<spmmGrid, 256, 0, stream>>>(T3, T2, vals, rows, cols, 2.0f);

    // Fused WMMA GEMM + bias + ReLU
    cheb_gemm_kernel<<<(BB * NN) / 32, 512, 0, stream>>>(x, T1, T2, T3, weights, bias, out);
}